// GNN1_17471926960722
// MI455X (gfx1250) — compile-verified
//
#include <hip/hip_runtime.h>
#include <hip/hip_bf16.h>
#include <math.h>

typedef float v2f __attribute__((ext_vector_type(2)));
typedef float v8f __attribute__((ext_vector_type(8)));

#define N_DRUG 846
#define DIM    256
#define KNB    128
#define APAD   34    // A panel row stride: conflict-free column reads, even (b64 aligned)
#define BPAD   34    // B panel COLUMN stride ([col][k] layout): b64 fragment reads
#define BN_EPS 1e-5f

__device__ __forceinline__ v8f zero8() {
  v8f z = {0.f, 0.f, 0.f, 0.f, 0.f, 0.f, 0.f, 0.f};
  return z;
}

// ---------------------------------------------------------------------------
// Kernel 1: per-drug neighbor aggregation (WMMA f32 16x16x4, prefetch-pipelined)
// ---------------------------------------------------------------------------
__global__ __launch_bounds__(512)
void gnn_neighbor_kernel(const int*   __restrict__ drug_name,
                         const int*   __restrict__ adj_tail,
                         const int*   __restrict__ adj_relation,
                         const float* __restrict__ drug_table,
                         const float* __restrict__ rela_table,
                         const float* __restrict__ ent_table,
                         const float* __restrict__ W1,
                         const float* __restrict__ b1,
                         const float* __restrict__ W2,
                         const float* __restrict__ b2,
                         float*       __restrict__ went)
{
  __shared__ float Apan[KNB * APAD];   // 17.4 KB : A panel [row 0..127][k 0..31]
  __shared__ float Bpan[DIM * BPAD];   // 34.8 KB : B panel column-major [col 0..255][k 0..31]
  __shared__ float s_demb[DIM];
  __shared__ int   s_rel[KNB];
  __shared__ int   s_tail[KNB];
  __shared__ float s_score[KNB];
  __shared__ float s_w[KNB];

  const int n    = blockIdx.x;
  const int tid  = threadIdx.x;
  const int lane = tid & 31;
  const int wid  = tid >> 5;          // 0..15
  const int mt   = wid & 7;           // M tile: rows [16*mt, 16*mt+16)
  const int nh   = wid >> 3;          // N half: cols [128*nh, 128*nh+128)

  if (tid < KNB) {
    s_rel[tid]  = adj_relation[n * KNB + tid];
    s_tail[tid] = adj_tail[n * KNB + tid];
  }
  const int dn = drug_name[n];
  if (tid < DIM) s_demb[tid] = drug_table[(size_t)dn * DIM + tid];
  __syncthreads();

  const int asel     = lane >> 4;                    // 0/1: which K pair
  const int arow     = mt * 16 + (lane & 15);        // A row this lane reads
  const int bcolbase = nh * 128 + (lane & 15);       // N column for tile 0
  const int rowbase  = mt * 16 + (asel << 3);        // C-layout row for vgpr r

  const size_t wbase = (size_t)n * DIM * DIM;

  // ---- W panel prefetch: 32 rows x 256 cols, pair-of-rows per thread ----
  float4 wlo[2], whi[2];
  auto prefW = [&](const float* __restrict__ Wp, int kt) {
    const float* base = Wp + wbase + (size_t)(kt * 32) * DIM;
#pragma unroll
    for (int i = 0; i < 2; ++i) {
      const int p  = tid + i * 512;        // 0..1023
      const int kp = p >> 6;               // k-row pair 0..15
      const int c4 = p & 63;               // float4 column 0..63
      wlo[i] = ((const float4*)(base + (size_t)(2 * kp)     * DIM))[c4];
      whi[i] = ((const float4*)(base + (size_t)(2 * kp + 1) * DIM))[c4];
    }
  };
  auto storeW = [&]() {
#pragma unroll
    for (int i = 0; i < 2; ++i) {
      const int p  = tid + i * 512;
      const int kp = p >> 6;
      const int c  = (p & 63) << 2;
      const float l[4] = {wlo[i].x, wlo[i].y, wlo[i].z, wlo[i].w};
      const float h[4] = {whi[i].x, whi[i].y, whi[i].z, whi[i].w};
#pragma unroll
      for (int j = 0; j < 4; ++j) {
        v2f v; v.x = l[j]; v.y = h[j];
        *(v2f*)&Bpan[(c + j) * BPAD + 2 * kp] = v;   // ds_store_b64
      }
    }
  };

  // ---- A panel prefetch (GEMM1): gather rela rows, scale by drug_emb ----
  float4 ra0, ra1;
  const int ar  = tid >> 2;            // 0..127
  const int ac0 = (tid & 3) * 8;       // 0,8,16,24
  auto prefA = [&](int kt) {
    const float* rr = rela_table + (size_t)s_rel[ar] * DIM + kt * 32 + ac0;
    ra0 = *(const float4*)rr;
    ra1 = *(const float4*)(rr + 4);
  };
  auto storeA = [&](int kt) {
    const float av[8] = {ra0.x, ra0.y, ra0.z, ra0.w, ra1.x, ra1.y, ra1.z, ra1.w};
#pragma unroll
    for (int j = 0; j < 4; ++j) {
      v2f v;
      v.x = av[2 * j]     * s_demb[kt * 32 + ac0 + 2 * j];
      v.y = av[2 * j + 1] * s_demb[kt * 32 + ac0 + 2 * j + 1];
      *(v2f*)&Apan[ar * APAD + ac0 + 2 * j] = v;     // ds_store_b64
    }
  };

  // ---- 64x WMMA over one staged panel ----
  auto mm = [&](v8f* acc) {
#pragma unroll
    for (int ks = 0; ks < 8; ++ks) {
      const int kc = ks * 4 + asel * 2;
      const v2f a = *(const v2f*)&Apan[arow * APAD + kc];          // ds_load_b64
#pragma unroll
      for (int t = 0; t < 8; ++t) {
        const v2f b = *(const v2f*)&Bpan[(bcolbase + t * 16) * BPAD + kc]; // ds_load_b64
        acc[t] = __builtin_amdgcn_wmma_f32_16x16x4_f32(
            false, a, false, b, (short)0, acc[t], false, false);
      }
    }
  };

  // ---------------- GEMM 1: drug_rel @ W1 ----------------
  v8f acc1[8];
#pragma unroll
  for (int t = 0; t < 8; ++t) acc1[t] = zero8();

  prefA(0);
  prefW(W1, 0);
  for (int kt = 0; kt < 8; ++kt) {
    __syncthreads();                 // prior panel reads complete
    storeA(kt);
    storeW();
    __syncthreads();                 // panel visible
    if (kt < 7) { prefA(kt + 1); prefW(W1, kt + 1); }   // overlap with compute
    mm(acc1);
  }

  prefW(W2, 0);                      // hide W2 panel-0 latency behind epilogue

  // bias + relu (C layout: vgpr r -> row rowbase+r, col bcolbase+16t)
#pragma unroll
  for (int t = 0; t < 8; ++t) {
    const int col = bcolbase + t * 16;
#pragma unroll
    for (int r = 0; r < 8; ++r) {
      float v = acc1[t][r] + b1[(rowbase + r) * DIM + col];
      acc1[t][r] = v > 0.f ? v : 0.f;
    }
  }

  // ---------------- GEMM 2: h1 @ W2 ----------------
  v8f acc2[8];
#pragma unroll
  for (int t = 0; t < 8; ++t) acc2[t] = zero8();

  for (int kt = 0; kt < 8; ++kt) {
    __syncthreads();
    // stage h1 columns [kt*32, kt*32+32) from owning half's C regs into A panel
    if (nh == (kt >> 2)) {
      const int t0 = (kt * 2) & 7;
#pragma unroll
      for (int tt = 0; tt < 2; ++tt) {
        const int t  = t0 + tt;
        const int cl = tt * 16 + (lane & 15);
#pragma unroll
        for (int r = 0; r < 8; ++r)
          Apan[(rowbase + r) * APAD + cl] = acc1[t][r];
      }
    }
    storeW();
    __syncthreads();
    if (kt < 7) prefW(W2, kt + 1);
    mm(acc2);
  }

  // ---------------- row sums (+b2) -> softmax weights ----------------
  float rp[8];
#pragma unroll
  for (int r = 0; r < 8; ++r) rp[r] = 0.f;
#pragma unroll
  for (int t = 0; t < 8; ++t) {
    const int col = bcolbase + t * 16;
#pragma unroll
    for (int r = 0; r < 8; ++r)
      rp[r] += acc2[t][r] + b2[(rowbase + r) * DIM + col];
  }
  if (tid < KNB) s_score[tid] = 0.f;
  __syncthreads();
#pragma unroll
  for (int r = 0; r < 8; ++r)
    atomicAdd(&s_score[rowbase + r], rp[r]);   // ds_add_f32
  __syncthreads();

  if (tid == 0) {
    float mx = -3.0e38f;
    for (int m = 0; m < KNB; ++m) mx = fmaxf(mx, s_score[m]);
    float sum = 0.f;
    for (int m = 0; m < KNB; ++m) {
      float e = __expf(s_score[m] - mx);
      s_w[m] = e;
      sum += e;
    }
    const float inv = 1.f / sum;
    for (int m = 0; m < KNB; ++m) s_w[m] *= inv;
  }
  __syncthreads();

  // ---------------- weighted entity aggregation ----------------
  {
    const int d    = tid & 255;
    const int half = tid >> 8;
    float s = 0.f;
    for (int m = half * 64; m < half * 64 + 64; ++m)
      s += s_w[m] * ent_table[(size_t)s_tail[m] * DIM + d];
    Apan[half * 256 + d] = s;     // Apan reused as scratch (post-barrier)
  }
  __syncthreads();
  if (tid < DIM)
    went[(size_t)n * DIM + tid] = Apan[tid] + Apan[256 + tid];
}

// ---------------------------------------------------------------------------
// Kernel 2: x = relu([went, drug_emb] @ lin_w^T + lin_b); accumulate BN stats
// ---------------------------------------------------------------------------
__global__ __launch_bounds__(256)
void lin_bnstat_kernel(const int*   __restrict__ drug_name,
                       const float* __restrict__ drug_table,
                       const float* __restrict__ went,
                       const float* __restrict__ lin_w,
                       const float* __restrict__ lin_b,
                       float*       __restrict__ xr,
                       float*       __restrict__ sums,
                       float*       __restrict__ sumsq)
{
  __shared__ float de[2 * DIM];
  const int n = blockIdx.x;
  const int e = threadIdx.x;
  de[e]       = went[(size_t)n * DIM + e];
  de[DIM + e] = drug_table[(size_t)drug_name[n] * DIM + e];
  __syncthreads();

  float acc = lin_b[e];
  const float* wrow = lin_w + (size_t)e * (2 * DIM);
#pragma unroll 8
  for (int j = 0; j < 2 * DIM; ++j) acc += de[j] * wrow[j];
  const float x = acc > 0.f ? acc : 0.f;
  xr[(size_t)n * DIM + e] = x;
  atomicAdd(&sums[e],  x);       // global_atomic_add_f32
  atomicAdd(&sumsq[e], x * x);
}

// ---------------------------------------------------------------------------
// Kernel 3: fold batch stats into scale/shift
// ---------------------------------------------------------------------------
__global__ __launch_bounds__(256)
void bn_fold_kernel(const float* __restrict__ sums,
                    const float* __restrict__ sumsq,
                    const float* __restrict__ gamma,
                    const float* __restrict__ beta,
                    float*       __restrict__ scsh)
{
  const int e = threadIdx.x;
  const float invN = 1.f / (float)N_DRUG;
  const float mean = sums[e] * invN;
  const float var  = sumsq[e] * invN - mean * mean;
  const float sc   = gamma[e] * rsqrtf(var + BN_EPS);
  scsh[e]       = sc;
  scsh[DIM + e] = beta[e] - mean * sc;
}

// ---------------------------------------------------------------------------
// Kernel 4: apply BN
// ---------------------------------------------------------------------------
__global__ __launch_bounds__(256)
void bn_apply_kernel(const float* __restrict__ xr,
                     const float* __restrict__ scsh,
                     float*       __restrict__ out)
{
  const int n = blockIdx.x;
  const int e = threadIdx.x;
  out[(size_t)n * DIM + e] = xr[(size_t)n * DIM + e] * scsh[e] + scsh[DIM + e];
}

// ---------------------------------------------------------------------------
extern "C" void kernel_launch(void* const* d_in, const int* in_sizes, int n_in,
                              void* d_out, int out_size, void* d_ws, size_t ws_size,
                              hipStream_t stream) {
  const int*   drug_name = (const int*)  d_in[0];
  const int*   adj_tail  = (const int*)  d_in[1];
  const int*   adj_rel   = (const int*)  d_in[2];
  const float* drug_tab  = (const float*)d_in[3];
  const float* rela_tab  = (const float*)d_in[4];
  const float* ent_tab   = (const float*)d_in[5];
  const float* W1        = (const float*)d_in[6];
  const float* b1        = (const float*)d_in[7];
  const float* W2        = (const float*)d_in[8];
  const float* b2        = (const float*)d_in[9];
  const float* lin_w     = (const float*)d_in[10];
  const float* lin_b     = (const float*)d_in[11];
  const float* bn_gamma  = (const float*)d_in[12];
  const float* bn_beta   = (const float*)d_in[13];

  char* ws = (char*)d_ws;
  const size_t NXD = (size_t)N_DRUG * DIM * sizeof(float);  // 866,304 B
  float* went  = (float*)(ws);
  float* xr    = (float*)(ws + NXD);
  float* sums  = (float*)(ws + 2 * NXD);
  float* sumsq = (float*)(ws + 2 * NXD + DIM * sizeof(float));
  float* scsh  = (float*)(ws + 2 * NXD + 2 * DIM * sizeof(float));

  hipMemsetAsync(sums, 0, 2 * DIM * sizeof(float), stream);

  gnn_neighbor_kernel<<<N_DRUG, 512, 0, stream>>>(
      drug_name, adj_tail, adj_rel, drug_tab, rela_tab, ent_tab,
      W1, b1, W2, b2, went);

  lin_bnstat_kernel<<<N_DRUG, 256, 0, stream>>>(
      drug_name, drug_tab, went, lin_w, lin_b, xr, sums, sumsq);

  bn_fold_kernel<<<1, 256, 0, stream>>>(sums, sumsq, bn_gamma, bn_beta, scsh);

  bn_apply_kernel<<<N_DRUG, 256, 0, stream>>>(xr, scsh, (float*)d_out);
}